// ISWAPLayer_2379411882435
// MI455X (gfx1250) — compile-verified
//
#include <hip/hip_runtime.h>
#include <cstdint>

// iSWAP on qubits (3,7) of a 12-qubit state, batch 2048.
// out_re[i] = swap ? x_im[i^0x110] : x_re[i]
// out_im[i] = swap ? -x_re[i^0x110] : x_im[i]   where swap = bit8(i) != bit4(i)
//
// Memory-bound permutation: 128 MB total traffic -> ~5.5 us at 23.3 TB/s.
// Strategy: stage contiguous 512-element-aligned windows in LDS via CDNA5
// async global->LDS b128 loads (ASYNCcnt path), permute in LDS, store
// coalesced b128 non-temporal.

#define THREADS 256
#define TILE 1024u   // floats per array per block (two 512 permutation windows)

typedef float f32x4 __attribute__((ext_vector_type(4)));
typedef int   i32x4 __attribute__((ext_vector_type(4)));
typedef __attribute__((address_space(1))) i32x4 g1_i32x4;   // global
typedef __attribute__((address_space(3))) i32x4 l3_i32x4;   // LDS

__device__ __forceinline__ void async_copy_b128(const float* gsrc, float* lds_dst) {
#if __has_builtin(__builtin_amdgcn_global_load_async_to_lds_b128)
    __builtin_amdgcn_global_load_async_to_lds_b128(
        (g1_i32x4*)(uintptr_t)gsrc,      // AS1: numeric == flat address
        (l3_i32x4*)lds_dst,              // AS3: addrspacecast strips aperture
        /*imm_off=*/0, /*cpol=*/0);
#else
    unsigned lds_off = (unsigned)(uintptr_t)(l3_i32x4*)lds_dst;
    asm volatile("global_load_async_to_lds_b128 %0, %1, off"
                 :: "v"(lds_off), "v"(gsrc) : "memory");
#endif
}

__device__ __forceinline__ void wait_async0() {
#if __has_builtin(__builtin_amdgcn_s_wait_asynccnt)
    __builtin_amdgcn_s_wait_asynccnt(0);
#else
    asm volatile("s_wait_asynccnt 0" ::: "memory");
#endif
}

__global__ __launch_bounds__(THREADS)
void iswap_permute_kernel(const float* __restrict__ x_re,
                          const float* __restrict__ x_im,
                          float* __restrict__ out,
                          unsigned n_elems /* per array */) {
    __shared__ float s_re[TILE];
    __shared__ float s_im[TILE];

    const unsigned t    = threadIdx.x;
    const unsigned base = blockIdx.x * TILE;   // 1024-aligned -> 512-window aligned
    const unsigned e0   = 4u * t;

    // Stage: contiguous 128-bit async copies, global -> LDS.
    async_copy_b128(x_re + base + e0, &s_re[e0]);
    async_copy_b128(x_im + base + e0, &s_im[e0]);
    wait_async0();        // this wave's async LDS writes landed
    __syncthreads();      // make all waves' tiles visible

    float* __restrict__ out_re = out;
    float* __restrict__ out_im = out + n_elems;

    f32x4 vr, vi;
#pragma unroll
    for (int k = 0; k < 4; ++k) {
        const unsigned e  = e0 + (unsigned)k;                 // tile-local index
        const unsigned sw = ((e >> 8) ^ (e >> 4)) & 1u;       // bit8 != bit4 ?
        const unsigned j  = e ^ (sw ? 0x110u : 0u);           // swap bits 8 and 4
        const float a = s_re[j];
        const float b = s_im[j];
        vr[k] = sw ?  b : a;    // real:  x_im[j]  or x_re[i]
        vi[k] = sw ? -a : b;    // imag: -x_re[j]  or x_im[i]
    }

    // Perfectly coalesced 16B stores; streaming data -> non-temporal.
    __builtin_nontemporal_store(vr, (f32x4*)(out_re + base + e0));
    __builtin_nontemporal_store(vi, (f32x4*)(out_im + base + e0));
}

extern "C" void kernel_launch(void* const* d_in, const int* in_sizes, int n_in,
                              void* d_out, int out_size, void* d_ws, size_t ws_size,
                              hipStream_t stream) {
    (void)n_in; (void)out_size; (void)d_ws; (void)ws_size;
    const float* x_re = (const float*)d_in[0];   // [2048, 4096, 1] f32
    const float* x_im = (const float*)d_in[1];   // [2048, 4096, 1] f32
    float* out = (float*)d_out;                  // [2, 2048, 4096, 1] f32

    const unsigned n_elems = (unsigned)in_sizes[0];     // 8,388,608
    const unsigned blocks  = n_elems / TILE;            // 8192

    iswap_permute_kernel<<<blocks, THREADS, 0, stream>>>(x_re, x_im, out, n_elems);
}